// PositionEncoding_33543694582279
// MI455X (gfx1250) — compile-verified
//
#include <hip/hip_runtime.h>
#include <stdint.h>

// PositionEncoding: out[b,s,d] = x[b,s,d] + pe[s,d]
//   pe[s,d] = (d even ? sin : cos)(s * 10000^(-d/1024))
// B=8, S=4096, D=1024, fp32.
//
// HBM-bound: 268 MB mandatory traffic -> ~11.5us floor @ 23.3 TB/s.
// Strategy: one block per s-row; PE row (4KB) computed once into LDS and
// shared by 8 waves (one wave per batch). Each wave streams its x row via
// CDNA5 async global->LDS b128 loads (ASYNCcnt, double-buffered, no VGPR
// landing zone, th:NT since x is read-once), adds PE from LDS, and writes
// out with non-temporal b128 stores. No WMMA: no matrix contraction here.

typedef float v4f __attribute__((ext_vector_type(4)));

#define S_LEN   4096
#define D_MODEL 1024
#define BATCH   8

// -log2(10000)/1024
#define NEG_L2W_OVER_D (-0.012976281620653759f)

__global__ __launch_bounds__(256) void pe_add_kernel(const float* __restrict__ x,
                                                     float* __restrict__ out)
{
    // Single __shared__ array => group-segment byte offset 0, so LDS byte
    // offsets for the async-to-LDS instruction are just float-index * 4.
    // Layout (floats):
    //   [0,    1024) : PE row for this s               (4 KB)
    //   [1024, 3072) : per-wave double buffers: wave w, buf p at
    //                  1024 + w*256 + p*128            (8 KB)
    __shared__ float lds[1024 + BATCH * 2 * 128];

    const int s    = blockIdx.x;          // one block per sequence position
    const int t    = threadIdx.x;         // 256 threads = 8 waves (wave32)
    const int w    = t >> 5;              // wave id == batch id
    const int lane = t & 31;

    // ---- Phase 1: PE row into LDS (thread t covers d = 4t .. 4t+3) ----
    // Accurate ocml sinf/cosf: angles reach 4095 rad; the hardware
    // v_sin_f32 fast path loses ~10 bits there. The Payne-Hanek slow path
    // (|a| >= 32768) is branched out at runtime (EXECZ skip) for our range.
    {
        const int   j0 = t << 2;
        const float fs = (float)s;
        const float a0 = fs * exp2f(NEG_L2W_OVER_D * (float)(j0 + 0));
        const float a1 = fs * exp2f(NEG_L2W_OVER_D * (float)(j0 + 1));
        const float a2 = fs * exp2f(NEG_L2W_OVER_D * (float)(j0 + 2));
        const float a3 = fs * exp2f(NEG_L2W_OVER_D * (float)(j0 + 3));
        v4f pe;
        pe.x = sinf(a0);   // even d -> sin
        pe.y = cosf(a1);   // odd  d -> cos
        pe.z = sinf(a2);
        pe.w = cosf(a3);
        *(v4f*)&lds[j0] = pe;             // ds_store_b128
    }
    __syncthreads();

    // ---- Phase 2: wave w streams batch b = w through LDS ----
    const size_t rowBase = ((size_t)w * S_LEN + (size_t)s) * D_MODEL;
    const v4f* __restrict__ xrow = (const v4f*)(x + rowBase);   // 256 x float4
    v4f*       __restrict__ orow = (v4f*)(out + rowBase);

    // LDS byte offsets for this wave's double buffers (chunk = 512 B/wave)
    const unsigned ldsXByte = 4096u + (unsigned)w * 1024u;
    const unsigned laneByte = (unsigned)lane * 16u;

    // Prologue: async-issue chunk 0 (lane l pulls 16B; 512B per instruction).
    // th:TH_LOAD_NT -> x is read exactly once; don't churn 134MB through L2.
    {
        const unsigned dst = ldsXByte + laneByte;            // buffer 0
        const v4f*     src = xrow + lane;
        asm volatile("global_load_async_to_lds_b128 %0, %1, off th:TH_LOAD_NT"
                     :: "v"(dst), "v"(src) : "memory");
    }

    // Steady state: issue i+1 into the other buffer, wait oldest, consume i.
    // Buffers are wave-private: no cross-wave barrier needed; the value-use
    // of chunk i's ds_load (DScnt wait) strictly precedes re-issuing into
    // the same buffer (chunk i+2) in the next iteration.
    #pragma unroll
    for (int i = 0; i < 8; ++i) {
        if (i + 1 < 8) {
            const unsigned dst = ldsXByte + ((unsigned)(i + 1) & 1u) * 512u + laneByte;
            const v4f*     src = xrow + (i + 1) * 32 + lane;
            asm volatile("global_load_async_to_lds_b128 %0, %1, off th:TH_LOAD_NT"
                         :: "v"(dst), "v"(src) : "memory");
            asm volatile("s_wait_asynccnt 1" ::: "memory");  // oldest chunk landed
        } else {
            asm volatile("s_wait_asynccnt 0" ::: "memory");  // last chunk landed
        }

        const v4f xv = *(const v4f*)&lds[1024 + w * 256 + (i & 1) * 128 + lane * 4];
        const v4f pv = *(const v4f*)&lds[i * 128 + lane * 4];
        const v4f o  = xv + pv;

        // out is write-once: keep it out of the cache hierarchy (th:NT)
        __builtin_nontemporal_store(o, orow + i * 32 + lane);
    }
}

extern "C" void kernel_launch(void* const* d_in, const int* in_sizes, int n_in,
                              void* d_out, int out_size, void* d_ws, size_t ws_size,
                              hipStream_t stream)
{
    (void)in_sizes; (void)n_in; (void)d_ws; (void)ws_size; (void)out_size;
    const float* x   = (const float*)d_in[0];
    float*       out = (float*)d_out;
    // One block per s position; 8 waves per block (one per batch).
    pe_add_kernel<<<S_LEN, 256, 0, stream>>>(x, out);
}